// CNN_GAT_66829691126432
// MI455X (gfx1250) — compile-verified
//
#include <hip/hip_runtime.h>
#include <cstddef>
#include <cstdint>

typedef float v2f __attribute__((ext_vector_type(2)));
typedef float v8f __attribute__((ext_vector_type(8)));

__device__ __forceinline__ v8f wmma_f32_16x16x4(v2f a, v2f b, v8f c) {
  // D = A(16x4 f32) * B(4x16 f32) + C(16x16 f32), wave32
  return __builtin_amdgcn_wmma_f32_16x16x4_f32(false, a, false, b, (short)0, c,
                                               false, false);
}

// CDNA5 async memory->LDS copy (ASYNCcnt-tracked), via inline asm so it works
// on both toolchains.  lds_addr = low 32 bits of generic LDS pointer.
__device__ __forceinline__ void async_load_b128(uint32_t lds_addr,
                                                const void* gaddr) {
  asm volatile("global_load_async_to_lds_b128 %0, %1, off"
               :: "v"(lds_addr), "v"(gaddr) : "memory");
}
__device__ __forceinline__ void wait_async0() {
  asm volatile("s_wait_asynccnt 0" ::: "memory");
}

// ---------------------------------------------------------------------------
// K1: conv1 (17->32, k=5, pad=2) + ReLU + MaxPool2.  x[128,17,4096] -> h1[128,32,2048]
// ---------------------------------------------------------------------------
__global__ __launch_bounds__(256) void k1_conv1(const float* __restrict__ x,
                                                const float* __restrict__ w,
                                                const float* __restrict__ bias,
                                                float* __restrict__ h1) {
  __shared__ float xs[17 * 260];
  __shared__ float wls[32 * 17 * 5];
  __shared__ float bls[32];
  const int b = blockIdx.x >> 4, chunk = blockIdx.x & 15;
  const int t0 = chunk * 256, tid = threadIdx.x;
  for (int i = tid; i < 17 * 260; i += 256) {
    int c = i / 260, tt = i - c * 260;
    int g = t0 + tt - 2;
    xs[i] = (g >= 0 && g < 4096) ? x[((size_t)b * 17 + c) * 4096 + g] : 0.f;
  }
  for (int i = tid; i < 32 * 17 * 5; i += 256) wls[i] = w[i];
  if (tid < 32) bls[tid] = bias[tid];
  __syncthreads();
  for (int idx = tid; idx < 32 * 128; idx += 256) {
    int oc = idx >> 7, p = idx & 127;
    float a0 = bls[oc], a1 = a0;
    const float* wp = wls + oc * 85;
    for (int c = 0; c < 17; ++c) {
      const float* xp = xs + c * 260 + 2 * p;
#pragma unroll
      for (int k = 0; k < 5; ++k) {
        float wv = wp[c * 5 + k];
        a0 = fmaf(wv, xp[k], a0);
        a1 = fmaf(wv, xp[k + 1], a1);
      }
    }
    h1[((size_t)b * 32 + oc) * 2048 + chunk * 128 + p] = fmaxf(fmaxf(a0, a1), 0.f);
  }
}

// ---------------------------------------------------------------------------
// K2: conv2 (32->64, k=5, pad=2) + ReLU + MaxPool2, output transposed.
//     h1[128,32,2048] -> h[128,1024,64]
// ---------------------------------------------------------------------------
__global__ __launch_bounds__(256) void k2_conv2(const float* __restrict__ h1,
                                                const float* __restrict__ w,
                                                const float* __restrict__ bias,
                                                float* __restrict__ h) {
  __shared__ float xs[32 * 260];
  __shared__ float wls[64 * 32 * 5];
  __shared__ float bls[64];
  const int b = blockIdx.x >> 3, chunk = blockIdx.x & 7;
  const int t0 = chunk * 256, tid = threadIdx.x;
  for (int i = tid; i < 32 * 260; i += 256) {
    int c = i / 260, tt = i - c * 260;
    int g = t0 + tt - 2;
    xs[i] = (g >= 0 && g < 2048) ? h1[((size_t)b * 32 + c) * 2048 + g] : 0.f;
  }
  for (int i = tid; i < 64 * 32 * 5; i += 256) wls[i] = w[i];
  if (tid < 64) bls[tid] = bias[tid];
  __syncthreads();
  for (int idx = tid; idx < 64 * 128; idx += 256) {
    int oc = idx & 63, p = idx >> 6;
    float a0 = bls[oc], a1 = a0;
    const float* wp = wls + oc * 160;
    for (int c = 0; c < 32; ++c) {
      const float* xp = xs + c * 260 + 2 * p;
#pragma unroll
      for (int k = 0; k < 5; ++k) {
        float wv = wp[c * 5 + k];
        a0 = fmaf(wv, xp[k], a0);
        a1 = fmaf(wv, xp[k + 1], a1);
      }
    }
    int n = chunk * 128 + p;
    h[((size_t)b * 1024 + n) * 64 + oc] = fmaxf(fmaxf(a0, a1), 0.f);
  }
}

// ---------------------------------------------------------------------------
// K3: per batch: hw = h @ gat_w^T  (WMMA f32 16x16x4, K=64), then
//     s_src/s_dst, S_max reduction, and the factorized attention terms:
//     E1, E2 (per source j), A, B, T (per dest i).  One block per batch.
// hwG is written PAIR-SWIZZLED for K4's B-operand:
//     hwG[b*16384 + (row>>1)*32 + col*2 + (row&1)]
// so (hw[2p][m], hw[2p+1][m]) are 8 contiguous bytes -> one ds_load_b64 in K4.
// ---------------------------------------------------------------------------
__global__ __launch_bounds__(256) void k3_gat_prep(
    const float* __restrict__ h, const float* __restrict__ gat_w,
    const float* __restrict__ att_src, const float* __restrict__ att_dst,
    float* __restrict__ hwG, float2* __restrict__ E12o,
    float2* __restrict__ ABo, float* __restrict__ To) {
  __shared__ float hwS[1024 * 16];   // 64 KB, standard layout for phase B
  __shared__ float wTs[64 * 16];     // gat_w transposed: wTs[k][o]
  __shared__ float attS[16], attD[16];
  __shared__ float red[256];
  const int b = blockIdx.x, tid = threadIdx.x;

  for (int i = tid; i < 1024; i += 256) wTs[i] = gat_w[(i & 15) * 64 + (i >> 4)];
  if (tid < 16) { attS[tid] = att_src[tid]; attD[tid] = att_dst[tid]; }
  __syncthreads();

  const int wave = tid >> 5, L = tid & 31, m = L & 15, half = L >> 4;
  // Preload B operand (gat_w^T) into registers, WMMA 4x16 K-slab layout:
  // v0 = rows {k, k+2}, v1 = rows {k+1, k+3} striped over lane halves.
  float2 bk[16];
#pragma unroll
  for (int k = 0; k < 16; ++k) {
    int kq = 4 * k + half * 2;
    bk[k].x = wTs[kq * 16 + m];
    bk[k].y = wTs[(kq + 1) * 16 + m];
  }
  for (int it = 0; it < 8; ++it) {
    int n0 = (wave * 8 + it) * 16;
    v8f acc = {0.f, 0.f, 0.f, 0.f, 0.f, 0.f, 0.f, 0.f};
    const float* abase = h + ((size_t)(b * 1024 + n0 + m)) * 64 + half * 2;
#pragma unroll
    for (int k = 0; k < 16; ++k) {
      v2f a = *(const v2f*)(abase + 4 * k);
      acc = wmma_f32_16x16x4(a, (v2f){bk[k].x, bk[k].y}, acc);
    }
#pragma unroll
    for (int vr = 0; vr < 8; ++vr) {
      int row = n0 + vr + half * 8;
      hwS[row * 16 + m] = acc[vr];
      hwG[(size_t)b * 16384 + (row >> 1) * 32 + m * 2 + (row & 1)] = acc[vr];
    }
  }
  __syncthreads();

  // s_src / s_dst + batch-wide max of s_src
  float ss[4], sd[4];
  float mx = -3.4e38f;
#pragma unroll
  for (int q = 0; q < 4; ++q) {
    int n = tid + 256 * q;
    float a0 = 0.f, a1 = 0.f;
#pragma unroll
    for (int o = 0; o < 16; ++o) {
      float v = hwS[n * 16 + o];
      a0 = fmaf(v, attS[o], a0);
      a1 = fmaf(v, attD[o], a1);
    }
    ss[q] = a0; sd[q] = a1;
    mx = fmaxf(mx, a0);
  }
  red[tid] = mx;
  __syncthreads();
  for (int s = 128; s > 0; s >>= 1) {
    if (tid < s) red[tid] = fmaxf(red[tid], red[tid + s]);
    __syncthreads();
  }
  const float smax = red[0];
#pragma unroll
  for (int q = 0; q < 4; ++q) {
    int n = tid + 256 * q;
    float e1 = expf(ss[q] - smax);
    float e2 = expf(0.2f * (ss[q] - smax));
    float dsm = sd[q] + smax;
    float Ai = expf(0.8f * fminf(dsm, 0.f));
    float Bi = expf(-0.8f * fmaxf(dsm, 0.f));
    float Ti = expf(-dsm);
    E12o[(size_t)b * 1024 + n] = make_float2(e1, e2);
    ABo[(size_t)b * 1024 + n]  = make_float2(Ai, Bi);
    To[(size_t)b * 1024 + n]   = Ti;
  }
}

// ---------------------------------------------------------------------------
// K4: attention aggregation.  Per wave: one 16-row i-tile; synthesize 16x4 w
// tiles in registers (exp-free) and WMMA against 4x16 hw tiles from LDS.
// LDS is filled with GLOBAL_LOAD_ASYNC_TO_LDS_B128 (ASYNCcnt path).
// Doubles w[i,i] to reproduce PyG's duplicated self-loop.  Emits per-tile
// column sums of num/denom for the mean pool.
// ---------------------------------------------------------------------------
__global__ __launch_bounds__(256) void k4_attn(
    const float* __restrict__ hwG, const float2* __restrict__ E12,
    const float2* __restrict__ AB, const float* __restrict__ Tg,
    float* __restrict__ partial) {
  __shared__ __align__(16) float hwL[1024 * 16];   // 64 KB, pair-swizzled
  __shared__ __align__(16) float2 e12L[1024];      // 8 KB
  const int b = blockIdx.x >> 3, seg = blockIdx.x & 7;
  const int tid = threadIdx.x;

  { // async staging: memory -> LDS directly, no VGPR round-trip
    const char* s4 = (const char*)(hwG + (size_t)b * 16384);
    const uint32_t lh = (uint32_t)(uintptr_t)(&hwL[0]);
#pragma unroll
    for (int q = 0; q < 16; ++q) {
      int i = tid + 256 * q;
      async_load_b128(lh + i * 16, s4 + (size_t)i * 16);
    }
    const char* e4 = (const char*)(E12 + (size_t)b * 1024);
    const uint32_t le = (uint32_t)(uintptr_t)(&e12L[0]);
#pragma unroll
    for (int q = 0; q < 2; ++q) {
      int i = tid + 256 * q;
      async_load_b128(le + i * 16, e4 + (size_t)i * 16);
    }
    wait_async0();
  }
  __syncthreads();

  const int wave = tid >> 5, L = tid & 31, m = L & 15, half = L >> 4;
  const int i0 = seg * 128 + wave * 16;
  const int irow = i0 + m;
  const float2 ab = AB[(size_t)b * 1024 + irow];
  const float thr = Tg[(size_t)b * 1024 + irow];
  const int jo = half * 2;

  v8f acc = {0.f, 0.f, 0.f, 0.f, 0.f, 0.f, 0.f, 0.f};
  float dsum = 0.f;
#pragma unroll 4
  for (int jb = 0; jb < 1024; jb += 4) {
    const int j0 = jb + jo;                        // even
    // E1/E2 for j0 and j0+1: 16 contiguous bytes -> one ds_load_b128
    const float4 e01 = *(const float4*)&e12L[j0];
    // w = (d_i + s_j >= 0) ? A_i*E1[j] : B_i*E2[j]; branch via E1 >= T_i
    float w0 = (e01.x >= thr) ? ab.x * e01.x : ab.y * e01.y;
    float w1 = (e01.z >= thr) ? ab.x * e01.z : ab.y * e01.w;
    if ((jb >> 4) == (i0 >> 4)) {      // wave-uniform: diag inside this slab
      if (j0 == irow) w0 += w0;        // duplicated self-loop -> count twice
      if (j0 + 1 == irow) w1 += w1;
    }
    dsum += w0 + w1;
    // pair-swizzled B operand: hw[j0][m], hw[j0+1][m] -> one ds_load_b64
    const float2 bp = *(const float2*)&hwL[(j0 >> 1) * 32 + m * 2];
    acc = wmma_f32_16x16x4((v2f){w0, w1}, (v2f){bp.x, bp.y}, acc);
  }
  // denom: combine the two lane-halves of each row, invert once
  dsum += __shfl_xor(dsum, 16, 32);
  float inv = 1.0f / dsum;             // lane L holds row (L&15)'s 1/denom
  float csum = 0.f;
#pragma unroll
  for (int vr = 0; vr < 8; ++vr) {
    int rm = vr + half * 8;            // tile-local row index of acc[vr]
    float invr = __shfl(inv, rm, 32);
    csum += acc[vr] * invr;            // accumulate column sums for mean pool
  }
  csum += __shfl_xor(csum, 16, 32);    // fold rows 8..15 half into 0..7 half
  if (L < 16) partial[((size_t)b * 64 + seg * 8 + wave) * 16 + L] = csum;
}

// ---------------------------------------------------------------------------
// K5: reduce the 64 per-tile partials, mean over nodes, +gat_b, FC head.
// ---------------------------------------------------------------------------
__global__ __launch_bounds__(32) void k5_head(const float* __restrict__ partial,
                                              const float* __restrict__ gat_b,
                                              const float* __restrict__ fc_w,
                                              const float* __restrict__ fc_b,
                                              float* __restrict__ out) {
  const int b = blockIdx.x;
  const int L = threadIdx.x & 31;
  const int col = L & 15;
  float s = 0.f;
  for (int t = 0; t < 64; ++t) s += partial[((size_t)b * 64 + t) * 16 + col];
  float pooled = s * (1.0f / 1024.0f) + gat_b[col];
  float o0 = pooled * fc_w[col];
  float o1 = pooled * fc_w[16 + col];
#pragma unroll
  for (int d = 8; d >= 1; d >>= 1) {
    o0 += __shfl_xor(o0, d, 32);
    o1 += __shfl_xor(o1, d, 32);
  }
  if (L == 0) {
    out[b * 2 + 0] = o0 + fc_b[0];
    out[b * 2 + 1] = o1 + fc_b[1];
  }
}

// ---------------------------------------------------------------------------
extern "C" void kernel_launch(void* const* d_in, const int* in_sizes, int n_in,
                              void* d_out, int out_size, void* d_ws, size_t ws_size,
                              hipStream_t stream) {
  const float* x       = (const float*)d_in[0];
  const float* conv1_w = (const float*)d_in[1];
  const float* conv1_b = (const float*)d_in[2];
  const float* conv2_w = (const float*)d_in[3];
  const float* conv2_b = (const float*)d_in[4];
  const float* gat_w   = (const float*)d_in[5];
  const float* att_src = (const float*)d_in[6];
  const float* att_dst = (const float*)d_in[7];
  const float* gat_b   = (const float*)d_in[8];
  const float* fc_w    = (const float*)d_in[9];
  const float* fc_b    = (const float*)d_in[10];
  float* out = (float*)d_out;

  char* ws = (char*)d_ws;
  // [0, 32MB): h1 during conv stage; recycled for GAT-side arrays afterwards.
  float*  h1   = (float*)ws;                       // 128*32*2048*4  = 33,554,432
  float*  h    = (float*)(ws + 33554432);          // 128*1024*64*4  = 33,554,432
  float*  hwG  = (float*)ws;                       //  8,388,608 (reuses h1)
  float2* E12  = (float2*)(ws + 8388608);          //  1,048,576
  float2* AB   = (float2*)(ws + 9437184);          //  1,048,576
  float*  To   = (float*)(ws + 10485760);          //    524,288
  float*  part = (float*)(ws + 11010048);          //    524,288
  (void)in_sizes; (void)n_in; (void)out_size; (void)ws_size;

  k1_conv1<<<128 * 16, 256, 0, stream>>>(x, conv1_w, conv1_b, h1);
  k2_conv2<<<128 * 8, 256, 0, stream>>>(h1, conv2_w, conv2_b, h);
  k3_gat_prep<<<128, 256, 0, stream>>>(h, gat_w, att_src, att_dst, hwG, E12, AB, To);
  k4_attn<<<128 * 8, 256, 0, stream>>>(hwG, E12, AB, To, part);
  k5_head<<<128, 32, 0, stream>>>(part, gat_b, fc_w, fc_b, out);
}